// AttentionConv_6322191860015
// MI455X (gfx1250) — compile-verified
//
#include <hip/hip_runtime.h>

typedef __attribute__((ext_vector_type(2))) float v2f;
typedef __attribute__((ext_vector_type(4))) float v4f;
typedef __attribute__((ext_vector_type(8))) float v8f;

#define D_IN  192   // c*p*p = 3*8*8
#define D_QKV 576   // 3 * D_IN
#define LDA   196   // padded LDS row stride in floats (196 % 64 == 4 -> bank spread, even -> b64 aligned)
#define KCH   48    // K / 4 WMMA steps
#define NTILES 36   // 576 / 16

// ---------------------------------------------------------------------------
// Kernel 1: per-block QKV GEMM.  qkv[m][e] = sum_k feat[m][k] * W[e][k] + bq[e]
// M = 8*64*64 = 32768 unique interior blocks, K = 192, N = 576.
// Workgroup = 128 threads (4 waves); each WG computes 64 M-rows x 576 cols.
// Wave w owns M-rows [w*16, w*16+16); all 4 waves share the staged B tile.
// B is double-buffered in LDS so next-tile global loads overlap the WMMAs;
// the qkv bias is staged in LDS once to kill the per-tile global-load stall.
// ---------------------------------------------------------------------------
__global__ __launch_bounds__(128)
void qkv_gemm_kernel(const float* __restrict__ x,    // [8][3][512][512]
                     const float* __restrict__ W,    // [576][192]
                     const float* __restrict__ bq,   // [576]
                     float* __restrict__ qkv)        // [32768][576]
{
    __shared__ float ldsA[64 * LDA];        // 64 M-rows x 192 K (padded)
    __shared__ float ldsB[2][16 * LDA];     // double-buffered 16 N-rows x 192 K
    __shared__ float ldsBias[D_QKV];

    const int tid  = threadIdx.x;
    const int lane = tid & 31;
    const int wv   = tid >> 5;
    const int m0   = blockIdx.x * 64;

    // ---- stage A: 64 rows x 192 floats (x blocks, d = c*64 + py*8 + px) ----
    #pragma unroll
    for (int it = 0; it < 24; ++it) {
        int idx = tid + it * 128;          // 3072 float4 slots
        int lm  = idx / 48;
        int d0  = (idx % 48) * 4;
        int m   = m0 + lm;
        int bb  = m >> 12;
        int ii  = (m >> 6) & 63;
        int jj  = m & 63;
        int ch  = d0 >> 6;
        int py  = (d0 >> 3) & 7;
        int px  = d0 & 7;                  // 0 or 4 -> 16B aligned
        const float* src = x + ((((size_t)bb * 3 + ch) * 512) + (ii * 8 + py)) * 512
                             + jj * 8 + px;
        *(v4f*)(&ldsA[lm * LDA + d0]) = *(const v4f*)src;
    }

    // ---- stage bias once ----
    #pragma unroll
    for (int it = 0; it < 5; ++it) {
        int idx = tid + it * 128;
        if (idx < D_QKV) ldsBias[idx] = bq[idx];
    }

    // ---- prologue: stage B tile 0 into buffer 0 ----
    #pragma unroll
    for (int it = 0; it < 6; ++it) {
        int idx = tid + it * 128;          // 768 float4 slots
        int nn  = idx / 48;
        int k0  = (idx % 48) * 4;
        *(v4f*)(&ldsB[0][nn * LDA + k0]) = *(const v4f*)(&W[nn * D_IN + k0]);
    }
    __syncthreads();

    // ---- cache this wave's 16x192 A tile as WMMA fragments in registers ----
    // 16x4 f32 A layout: lanes 0-15 M=lane K={0,1}; lanes 16-31 M=lane-16 K={2,3}
    v2f afrag[KCH];
    {
        int row  = wv * 16 + (lane & 15);
        int kofs = (lane >> 4) * 2;
        #pragma unroll
        for (int kc = 0; kc < KCH; ++kc)
            afrag[kc] = *(const v2f*)(&ldsA[row * LDA + kc * 4 + kofs]);
    }

    const int brow = lane & 15;            // N column within tile
    const int kofs = (lane >> 4) * 2;      // K pair select
    const int col0 = lane & 15;
    const int mBase = m0 + wv * 16 + ((lane >> 4) << 3);

    for (int n = 0; n < NTILES; ++n) {
        const int cur = n & 1;

        // ---- issue next tile's global loads early (overlap with WMMAs) ----
        v4f breg[6];
        if (n + 1 < NTILES) {
            #pragma unroll
            for (int it = 0; it < 6; ++it) {
                int idx = tid + it * 128;
                int nn  = idx / 48;
                int k0  = (idx % 48) * 4;
                breg[it] = *(const v4f*)(&W[(size_t)((n + 1) * 16 + nn) * D_IN + k0]);
            }
        }

        // accumulator initialized with the fused bias (same for every M row)
        float bias = ldsBias[n * 16 + col0];
        v8f acc;
        #pragma unroll
        for (int r = 0; r < 8; ++r) acc[r] = bias;

        const float* Bbuf = &ldsB[cur][0];
        #pragma unroll
        for (int kc = 0; kc < KCH; ++kc) {
            v2f bfrag = *(const v2f*)(&Bbuf[brow * LDA + kc * 4 + kofs]);
            acc = __builtin_amdgcn_wmma_f32_16x16x4_f32(
                false, afrag[kc], false, bfrag, (short)0, acc, false, false);
        }

        // ---- park next tile into the other LDS buffer ----
        if (n + 1 < NTILES) {
            float* Bnxt = &ldsB[cur ^ 1][0];
            #pragma unroll
            for (int it = 0; it < 6; ++it) {
                int idx = tid + it * 128;
                int nn  = idx / 48;
                int k0  = (idx % 48) * 4;
                *(v4f*)(&Bnxt[nn * LDA + k0]) = breg[it];
            }
        }

        // ---- store D: VGPR r -> M=r (lanes 0-15) / M=r+8 (lanes 16-31) ----
        int col = n * 16 + col0;
        #pragma unroll
        for (int r = 0; r < 8; ++r)
            qkv[(size_t)(mBase + r) * D_QKV + col] = acc[r];

        __syncthreads();   // nxt buffer fully written before next iter reads it
    }
}

// ---------------------------------------------------------------------------
// Kernel 2: out[b,c,i*8+py,j*8+px] = q_center * sum_n k_n * v_n * bias[n,py,px]
// Border neighbors fall in the zero-padded ring -> their qkv is just b_qkv.
// bias[n](r,c) = exp(-(ay+ax)/4), ay/ax from neighbor direction (verified
// against the rot90/flip construction in _rel_bias).
// ---------------------------------------------------------------------------
__global__ __launch_bounds__(192)
void combine_kernel(const float* __restrict__ qkv,  // [32768][576]
                    const float* __restrict__ bq,   // [576]
                    float* __restrict__ out)        // [8][3][512][512]
{
    const int t   = threadIdx.x;           // d = ch*64 + py*8 + px
    const int blk = blockIdx.x;            // (b,i,j) interior block
    const int jj  = blk & 63;
    const int ii  = (blk >> 6) & 63;
    const int bb  = blk >> 12;

    const int py = (t >> 3) & 7;
    const int px = t & 7;

    float q = qkv[(size_t)blk * D_QKV + t];   // QK_SCALE == 1.0

    float s = 0.f;
    #pragma unroll
    for (int n = 0; n < 9; ++n) {
        int dy = n / 3 - 1, dx = n % 3 - 1;
        int ni = ii + dy, nj = jj + dx;
        float kk, vv;
        if (ni >= 0 && ni < 64 && nj >= 0 && nj < 64) {
            size_t mN = ((size_t)bb * 64 + ni) * 64 + nj;
            kk = qkv[mN * D_QKV + 192 + t];
            vv = qkv[mN * D_QKV + 384 + t];
        } else {
            kk = bq[192 + t];
            vv = bq[384 + t];
        }
        int ay = (dy < 0) ? (7 - py) : (dy > 0 ? py : 0);
        int ax = (dx < 0) ? (7 - px) : (dx > 0 ? px : 0);
        float bias = __expf(-(float)(ay + ax) * 0.25f);
        s += kk * vv * bias;
    }

    int ch = t >> 6;
    out[((((size_t)bb * 3 + ch) * 512) + (ii * 8 + py)) * 512 + jj * 8 + px] = q * s;
}

extern "C" void kernel_launch(void* const* d_in, const int* in_sizes, int n_in,
                              void* d_out, int out_size, void* d_ws, size_t ws_size,
                              hipStream_t stream) {
    const float* x  = (const float*)d_in[0];   // 8*3*512*512
    const float* W  = (const float*)d_in[1];   // 576*192
    const float* bq = (const float*)d_in[2];   // 576
    float* out = (float*)d_out;                // 8*3*512*512
    float* qkv = (float*)d_ws;                 // needs 32768*576*4 = 75.5 MB

    (void)in_sizes; (void)n_in; (void)out_size; (void)ws_size;

    // 32768 M-rows / 64 per workgroup = 512 workgroups of 128 threads
    qkv_gemm_kernel<<<512, 128, 0, stream>>>(x, W, bq, qkv);
    // one 192-thread block per output block
    combine_kernel<<<32768, 192, 0, stream>>>(qkv, bq, out);
}